// DGDAGRNN_75428215653096
// MI455X (gfx1250) — compile-verified
//
#include <hip/hip_runtime.h>
#include <hip/hip_bf16.h>

// ---------------- problem constants ----------------
constexpr int VHS    = 100;   // node hidden size
constexpr int VH3    = 300;   // 3*VHS (GRU gate width)
constexpr int CHS    = 30;    // classifier hidden
constexpr int NVT    = 3;     // vertex one-hot size
constexpr int LAYERS = 8;     // num_layers (static python scalar)
constexpr int ROUNDS = 2;     // nrounds   (static python scalar)
constexpr int KSTEPS = VHS / 4;  // 25 exact k-steps for 16x16x4 f32 WMMA
constexpr int NT1    = 7;     // col tiles for VHS (7*16 = 112)
constexpr int NT3    = 19;    // col tiles for VH3 (19*16 = 304)

typedef __attribute__((ext_vector_type(2))) float v2f;
typedef __attribute__((ext_vector_type(8))) float v8f;

__device__ __forceinline__ float sigmoidf_(float v) {
    return 1.0f / (1.0f + __expf(-v));
}

// ---------------- utility kernels ----------------
__global__ void zero_f32_kernel(float* p, int n) {
    int i = blockIdx.x * blockDim.x + threadIdx.x;
    if (i < n) p[i] = 0.0f;
}
__global__ void zero_i32_kernel(int* p, int n) {
    int i = blockIdx.x * blockDim.x + threadIdx.x;
    if (i < n) p[i] = 0;
}

// Pre-pack B operand into per-lane WMMA fragment order so the hot GEMM loops
// do one coalesced b64 load per fragment:
//   Bp[((ks*ntiles + t)*32 + lane)*2 + e] = W[c][kk],
//   kk = ks*4 + 2*(lane>>4) + e,  c = t*16 + (lane&15)   (zero-padded c>=rows)
// W is [rows][VHS] row-major (used as h @ W^T).
__global__ void pack_b_kernel(const float* __restrict__ W, float* __restrict__ Bp,
                              int rows, int ntiles) {
    int idx = blockIdx.x * blockDim.x + threadIdx.x;
    int total = KSTEPS * ntiles * 64;
    if (idx >= total) return;
    int e    = idx & 1;
    int lane = (idx >> 1) & 31;
    int rem  = idx >> 6;            // ks*ntiles + t
    int t    = rem % ntiles;
    int ks   = rem / ntiles;
    int kk   = ks * 4 + 2 * (lane >> 4) + e;
    int c    = t * 16 + (lane & 15);
    Bp[idx] = (c < rows) ? W[c * VHS + kk] : 0.0f;
}

// ---------------- node bucketing by layer (static across rounds) ----------------
__global__ void count_nodes_kernel(const int* __restrict__ nl, int* counts, int n) {
    int i = blockIdx.x * blockDim.x + threadIdx.x;
    if (i < n) atomicAdd(&counts[nl[i]], 1);
}
__global__ void scan_offsets_kernel(const int* __restrict__ counts, int* offs, int* cursor) {
    if (blockIdx.x == 0 && threadIdx.x == 0) {
        int s = 0;
        for (int l = 0; l < LAYERS; ++l) {
            offs[l] = s;
            s += counts[l];
            cursor[l] = 0;
        }
    }
}
__global__ void fill_lists_kernel(const int* __restrict__ nl, const int* __restrict__ offs,
                                  int* cursor, int* list, int n) {
    int i = blockIdx.x * blockDim.x + threadIdx.x;
    if (i < n) {
        int l = nl[i];
        int p = atomicAdd(&cursor[l], 1);
        list[offs[l] + p] = i;
    }
}

// ---------------- gn = sigmoid(h*Wg^T+bg) * (h*Wm^T), all N nodes ----------------
// One wave per 16-node tile; 7 col tiles x 2 GEMMs in registers; B pre-packed
// fragments (1 coalesced b64/lane each); D staged through LDS -> b128 stores.
__global__ void __launch_bounds__(32)
gn_gemm_kernel(const float* __restrict__ h,
               const float* __restrict__ BgP,   // packed [25][7][32][2]
               const float* __restrict__ BmP,
               const float* __restrict__ bg,
               float* __restrict__ gn) {
    const int lane = threadIdx.x;
    const int mrow = lane & 15;   // A: row M ; B/D: col N
    const int half = lane >> 4;   // selects K pair (A/B) and M+8 (D)
    const long n0 = (long)blockIdx.x * 16;

    v8f accG[NT1] = {};
    v8f accM[NT1] = {};
    const float* arow = h + (n0 + mrow) * VHS;
    const v2f* Bg = (const v2f*)BgP + lane;   // fragment stride = 32 v2f
    const v2f* Bm = (const v2f*)BmP + lane;

    for (int ks = 0; ks < KSTEPS; ++ks) {
        const int kk = ks * 4 + 2 * half;
        const v2f a = *(const v2f*)(arow + kk);
#pragma unroll
        for (int t = 0; t < NT1; ++t) {
            const v2f bgv = Bg[(ks * NT1 + t) * 32];
            const v2f bmv = Bm[(ks * NT1 + t) * 32];
            accG[t] = __builtin_amdgcn_wmma_f32_16x16x4_f32(
                false, a, false, bgv, (short)0, accG[t], false, false);
            accM[t] = __builtin_amdgcn_wmma_f32_16x16x4_f32(
                false, a, false, bmv, (short)0, accM[t], false, false);
        }
    }

    // elementwise sigmoid*map in the (identical) D layouts, stage via LDS so
    // global stores are coalesced b128 instead of 56 scattered b32 per lane
    __shared__ float lgn[16 * VHS];
#pragma unroll
    for (int t = 0; t < NT1; ++t) {
        const int c = t * 16 + mrow;
        if (c < VHS) {
            const float bgc = bg[c];
#pragma unroll
            for (int v = 0; v < 8; ++v) {
                const int row = v + 8 * half;
                lgn[row * VHS + c] = sigmoidf_(accG[t][v] + bgc) * accM[t][v];
            }
        }
    }
    __syncthreads();
    float4* dst = (float4*)(gn + n0 * VHS);          // 400B rows -> 16B aligned
    const float4* src = (const float4*)lgn;
    for (int idx = lane; idx < 16 * VHS / 4; idx += 32)
        dst[idx] = src[idx];
}

// ---------------- msg[dst] += gn[src] for edges with node_layer[dst]==l ----------------
// One wave per edge: predicate is wave-uniform, no divergence inside the wave.
__global__ void scatter_edges_kernel(const float* __restrict__ gn,
                                     const int* __restrict__ esrc,
                                     const int* __restrict__ edst,
                                     const int* __restrict__ nlayer,
                                     int l, float* __restrict__ msg, int E) {
    const int wave = (int)((blockIdx.x * blockDim.x + threadIdx.x) >> 5);
    const int lane = threadIdx.x & 31;
    if (wave >= E) return;
    const int dst = edst[wave];
    if (nlayer[dst] != l) return;
    const int src = esrc[wave];
    const float* g = gn + (long)src * VHS;
    float* m = msg + (long)dst * VHS;
    for (int f = lane; f < VHS; f += 32)
        atomicAdd(&m[f], g[f]);
}

// ---------------- GRU on the compacted layer-l node list ----------------
// gh = msg*Whh^T + bhh (19 col tiles, WMMA) staged through LDS, then gate math
// with gi recomputed on the fly (K=NVT=3) from LDS-staged W_ih/b_ih.
__global__ void __launch_bounds__(32)
gru_fused_kernel(const float* __restrict__ msg,
                 const float* __restrict__ BhhP,  // packed [25][19][32][2]
                 const float* __restrict__ bhh,
                 const float* __restrict__ x,     // [N][NVT]
                 const float* __restrict__ Wih,   // [VH3][NVT]
                 const float* __restrict__ bih,
                 const int* __restrict__ list,
                 const int* __restrict__ offs,
                 const int* __restrict__ counts,
                 int l, float* __restrict__ h) {
    const int cnt = counts[l];          // scalar load; uniform
    const int base = blockIdx.x * 16;
    if (base >= cnt) return;            // uniform exit, EXEC stays all-ones
    const int off = offs[l];

    const int lane = threadIdx.x;
    const int mrow = lane & 15;
    const int half = lane >> 4;

    int i = base + mrow;
    if (i > cnt - 1) i = cnt - 1;       // clamp (dup rows; stores are guarded)
    const int node = list[off + i];
    const float* arow = msg + (long)node * VHS;
    const v2f* Bh = (const v2f*)BhhP + lane;

    v8f acc[NT3] = {};
    for (int ks = 0; ks < KSTEPS; ++ks) {
        const int kk = ks * 4 + 2 * half;
        const v2f a = *(const v2f*)(arow + kk);
#pragma unroll
        for (int t = 0; t < NT3; ++t) {
            const v2f b = Bh[(ks * NT3 + t) * 32];
            acc[t] = __builtin_amdgcn_wmma_f32_16x16x4_f32(
                false, a, false, b, (short)0, acc[t], false, false);
        }
    }

    // stage gh tile (+ bias) and the tiny W_ih/b_ih tables into LDS
    __shared__ float lgh[16 * (NT3 * 16)];   // 16 x 304
    __shared__ float lWih[VH3 * NVT];        // 900
    __shared__ float lbih[VH3];              // 300
#pragma unroll
    for (int t = 0; t < NT3; ++t) {
        const int c = t * 16 + mrow;
        const float bb = (c < VH3) ? bhh[c] : 0.0f;
#pragma unroll
        for (int v = 0; v < 8; ++v) {
            lgh[(v + 8 * half) * (NT3 * 16) + c] = acc[t][v] + bb;
        }
    }
    for (int idx = lane; idx < VH3 * NVT; idx += 32) lWih[idx] = Wih[idx];
    for (int idx = lane; idx < VH3; idx += 32) lbih[idx] = bih[idx];
    __syncthreads();

    // gate phase: 16 rows x 100 features distributed over 32 lanes
    for (int idx = lane; idx < 16 * VHS; idx += 32) {
        const int m = idx / VHS;
        const int j = idx - m * VHS;
        const int i2 = base + m;
        if (i2 >= cnt) continue;
        const int nd = list[off + i2];

        const float x0 = x[nd * NVT + 0];
        const float x1 = x[nd * NVT + 1];
        const float x2 = x[nd * NVT + 2];

        const int jr = j, jz = VHS + j, jn = 2 * VHS + j;
        const float gi_r = lbih[jr] + x0 * lWih[jr * NVT + 0] + x1 * lWih[jr * NVT + 1] + x2 * lWih[jr * NVT + 2];
        const float gi_z = lbih[jz] + x0 * lWih[jz * NVT + 0] + x1 * lWih[jz * NVT + 1] + x2 * lWih[jz * NVT + 2];
        const float gi_n = lbih[jn] + x0 * lWih[jn * NVT + 0] + x1 * lWih[jn * NVT + 1] + x2 * lWih[jn * NVT + 2];

        const float gh_r = lgh[m * (NT3 * 16) + j];
        const float gh_z = lgh[m * (NT3 * 16) + VHS + j];
        const float gh_n = lgh[m * (NT3 * 16) + 2 * VHS + j];

        const float r = sigmoidf_(gi_r + gh_r);
        const float z = sigmoidf_(gi_z + gh_z);
        const float c = tanhf(gi_n + r * gh_n);
        const float hm = msg[(long)nd * VHS + j];   // GRU hidden input = msg
        h[(long)nd * VHS + j] = (1.0f - z) * c + z * hm;
    }
}

// ---------------- classifier head: sigmoid(W_c2 * relu(W_c1*h + b_c1) + b_c2) ----------------
__global__ void classifier_kernel(const float* __restrict__ h,
                                  const float* __restrict__ Wc1,  // [CHS][VHS]
                                  const float* __restrict__ bc1,
                                  const float* __restrict__ Wc2,  // [1][CHS]
                                  const float* __restrict__ bc2,
                                  float* __restrict__ out, int n) {
    int i = blockIdx.x * blockDim.x + threadIdx.x;
    if (i >= n) return;
    const float4* hr = (const float4*)(h + (long)i * VHS);   // 400B rows, 16B aligned
    float acc2 = bc2[0];
    for (int o = 0; o < CHS; ++o) {
        const float4* w = (const float4*)(Wc1 + o * VHS);
        float s = bc1[o];
        for (int k = 0; k < VHS / 4; ++k) {
            float4 hv = hr[k];
            float4 wv = w[k];
            s += hv.x * wv.x + hv.y * wv.y + hv.z * wv.z + hv.w * wv.w;
        }
        acc2 += fmaxf(s, 0.0f) * Wc2[o];
    }
    out[i] = sigmoidf_(acc2);
}

// ---------------- host driver ----------------
extern "C" void kernel_launch(void* const* d_in, const int* in_sizes, int n_in,
                              void* d_out, int out_size, void* d_ws, size_t ws_size,
                              hipStream_t stream) {
    const float* x   = (const float*)d_in[0];
    const float* Wih = (const float*)d_in[1];
    const float* bih = (const float*)d_in[2];
    const float* Whh = (const float*)d_in[3];
    const float* bhh = (const float*)d_in[4];
    const float* Wg  = (const float*)d_in[5];
    const float* bg  = (const float*)d_in[6];
    const float* Wm  = (const float*)d_in[7];
    const float* Wc1 = (const float*)d_in[8];
    const float* bc1 = (const float*)d_in[9];
    const float* Wc2 = (const float*)d_in[10];
    const float* bc2 = (const float*)d_in[11];
    const int* nl    = (const int*)d_in[12];
    const int* esrc  = (const int*)d_in[13];
    const int* edst  = (const int*)d_in[14];

    const int N = in_sizes[0] / NVT;   // 65536
    const int E = in_sizes[13];        // 262144

    // carve workspace (~80 MB), 256B aligned slabs
    char* w = (char*)d_ws;
    auto alloc = [&](size_t bytes) -> void* {
        void* p = (void*)w;
        w += (bytes + 255) & ~(size_t)255;
        return p;
    };
    float* h    = (float*)alloc((size_t)N * VHS * sizeof(float));
    float* msg  = (float*)alloc((size_t)N * VHS * sizeof(float));
    float* gn   = (float*)alloc((size_t)N * VHS * sizeof(float));
    float* BgP  = (float*)alloc((size_t)KSTEPS * NT1 * 64 * sizeof(float));
    float* BmP  = (float*)alloc((size_t)KSTEPS * NT1 * 64 * sizeof(float));
    float* BhhP = (float*)alloc((size_t)KSTEPS * NT3 * 64 * sizeof(float));
    int* counts = (int*)alloc(LAYERS * sizeof(int));
    int* offs   = (int*)alloc(LAYERS * sizeof(int));
    int* cursor = (int*)alloc(LAYERS * sizeof(int));
    int* list   = (int*)alloc((size_t)N * sizeof(int));

    const int nv = N * VHS;

    // init: h = 0, bucket nodes by layer, pack B operands
    zero_f32_kernel<<<(nv + 255) / 256, 256, 0, stream>>>(h, nv);
    zero_i32_kernel<<<1, 32, 0, stream>>>(counts, LAYERS);
    pack_b_kernel<<<(KSTEPS * NT1 * 64 + 255) / 256, 256, 0, stream>>>(Wg, BgP, VHS, NT1);
    pack_b_kernel<<<(KSTEPS * NT1 * 64 + 255) / 256, 256, 0, stream>>>(Wm, BmP, VHS, NT1);
    pack_b_kernel<<<(KSTEPS * NT3 * 64 + 255) / 256, 256, 0, stream>>>(Whh, BhhP, VH3, NT3);
    count_nodes_kernel<<<(N + 255) / 256, 256, 0, stream>>>(nl, counts, N);
    scan_offsets_kernel<<<1, 1, 0, stream>>>(counts, offs, cursor);
    fill_lists_kernel<<<(N + 255) / 256, 256, 0, stream>>>(nl, offs, cursor, list, N);

    const int ntiles = (N + 15) / 16;
    for (int r = 0; r < ROUNDS; ++r) {
        for (int l = 0; l < LAYERS; ++l) {
            gn_gemm_kernel<<<ntiles, 32, 0, stream>>>(h, BgP, BmP, bg, gn);
            zero_f32_kernel<<<(nv + 255) / 256, 256, 0, stream>>>(msg, nv);
            scatter_edges_kernel<<<(E + 7) / 8, 256, 0, stream>>>(gn, esrc, edst, nl, l, msg, E);
            gru_fused_kernel<<<ntiles, 32, 0, stream>>>(msg, BhhP, bhh, x, Wih, bih,
                                                        list, offs, counts, l, h);
        }
    }
    classifier_kernel<<<(N + 255) / 256, 256, 0, stream>>>(h, Wc1, bc1, Wc2, bc2,
                                                           (float*)d_out, N);
}